// MutualConsistency_67336497267227
// MI455X (gfx1250) — compile-verified
//
#include <hip/hip_runtime.h>
#include <hip/hip_bf16.h>

typedef __attribute__((ext_vector_type(2))) float v2f;
typedef __attribute__((ext_vector_type(8))) float v8f;

#define NB_DICE 512

// ---------------------------------------------------------------------------
// Stage 1: five global reductions over the three masks (bandwidth bound).
//   s0 = sum(classic*gt)  s1 = sum(classic)  s2 = sum(gt)
//   s3 = sum(classic*snake)  s4 = sum(snake)
// ---------------------------------------------------------------------------
__global__ __launch_bounds__(256)
void mc_dice_partials(const float4* __restrict__ gt,
                      const float4* __restrict__ snake,
                      const float4* __restrict__ classic,
                      float* __restrict__ partial, int n4) {
  float a0 = 0.f, a1 = 0.f, a2 = 0.f, a3 = 0.f, a4 = 0.f;
  for (int i = blockIdx.x * blockDim.x + threadIdx.x; i < n4;
       i += gridDim.x * blockDim.x) {
    float4 g = gt[i];
    float4 s = snake[i];
    float4 c = classic[i];
    a0 += c.x * g.x + c.y * g.y + c.z * g.z + c.w * g.w;
    a1 += c.x + c.y + c.z + c.w;
    a2 += g.x + g.y + g.z + g.w;
    a3 += c.x * s.x + c.y * s.y + c.z * s.z + c.w * s.w;
    a4 += s.x + s.y + s.z + s.w;
  }
  __shared__ float red[5][256];
  int tid = threadIdx.x;
  red[0][tid] = a0; red[1][tid] = a1; red[2][tid] = a2;
  red[3][tid] = a3; red[4][tid] = a4;
  __syncthreads();
  for (int off = 128; off; off >>= 1) {
    if (tid < off) {
      for (int s = 0; s < 5; ++s) red[s][tid] += red[s][tid + off];
    }
    __syncthreads();
  }
  if (tid == 0) {
    for (int s = 0; s < 5; ++s) partial[s * NB_DICE + blockIdx.x] = red[s][0];
  }
}

// ---------------------------------------------------------------------------
// Stage 2: circular cross-correlation via V_WMMA_F32_16X16X4_F32.
// blockIdx.x encodes (term[2], batch[16], vtile[4]); one wave per block.
// corr[u + 16*v] = sum_k p_flat[(k + 2u) mod 2048] * q_flat[(k - 32v) mod 2048]
// Both vectors are stored twice in LDS so all reads are wrap-free
// (ds_load_b64 with immediate offsets, no per-step mask/shift VALU).
// Four independent accumulator chains over interleaved K slices.
// ---------------------------------------------------------------------------
__global__ __launch_bounds__(32)
void mc_rot_corr(const float* __restrict__ predA, const float* __restrict__ refA,
                 const float* __restrict__ predB, const float* __restrict__ refB,
                 float* __restrict__ mc, float* __restrict__ ssq) {
  __shared__ float4 sp4[1024];   // p_flat duplicated: 4096 floats
  __shared__ float4 sq4[1024];   // q_flat duplicated: 4096 floats

  int blk  = blockIdx.x;
  int term = blk >> 6;          // 0..1
  int b    = (blk >> 2) & 15;   // 0..15
  int t    = blk & 3;           // 0..3  (v-tile)

  const float* pred = term ? predB : predA;
  const float* ref  = term ? refB  : refA;
  const float4* p4 = (const float4*)(pred + b * 2048);
  const float4* q4 = (const float4*)(ref  + b * 2048);

  int lane = threadIdx.x;
  float sum_p2 = 0.f, sum_q2 = 0.f;
  for (int i = lane; i < 512; i += 32) {
    float4 p = p4[i];
    float4 q = q4[i];
    sp4[i] = p; sp4[i + 512] = p;
    sq4[i] = q; sq4[i + 512] = q;
    sum_p2 += p.x * p.x + p.y * p.y + p.z * p.z + p.w * p.w;
    sum_q2 += q.x * q.x + q.y * q.y + q.z * q.z + q.w * q.w;
  }
  __syncthreads();

  int half = lane >> 4;  // which K-pair this lane supplies within a WMMA
  // A[u,k] = p[(k + 2u) mod 2048]: lane base = 2u + 2*half   (u = lane&15)
  // B[k,v] = q[(k - 32v) mod 2048]: lane base = (-32v mod 2048) + 2*half
  int aoff = 2 * (lane & 15) + 2 * half;                              // <= 32
  int boff = ((-32 * (16 * t + (lane & 15))) & 2047) + 2 * half;      // <= 2018
  const float* pa = (const float*)sp4 + aoff;  // max read idx 2077 < 4096
  const float* pb = (const float*)sq4 + boff;  // max read idx 4063 < 4096

  const v8f z = {0.f, 0.f, 0.f, 0.f, 0.f, 0.f, 0.f, 0.f};
  v8f acc0 = z, acc1 = z, acc2 = z, acc3 = z;
#pragma unroll 8
  for (int kb = 0; kb < 2048; kb += 16) {
    v2f a0 = *(const v2f*)(pa + kb);
    v2f b0 = *(const v2f*)(pb + kb);
    v2f a1 = *(const v2f*)(pa + kb + 4);
    v2f b1 = *(const v2f*)(pb + kb + 4);
    v2f a2 = *(const v2f*)(pa + kb + 8);
    v2f b2 = *(const v2f*)(pb + kb + 8);
    v2f a3 = *(const v2f*)(pa + kb + 12);
    v2f b3 = *(const v2f*)(pb + kb + 12);
    acc0 = __builtin_amdgcn_wmma_f32_16x16x4_f32(false, a0, false, b0,
                                                 (short)0, acc0, false, false);
    acc1 = __builtin_amdgcn_wmma_f32_16x16x4_f32(false, a1, false, b1,
                                                 (short)0, acc1, false, false);
    acc2 = __builtin_amdgcn_wmma_f32_16x16x4_f32(false, a2, false, b2,
                                                 (short)0, acc2, false, false);
    acc3 = __builtin_amdgcn_wmma_f32_16x16x4_f32(false, a3, false, b3,
                                                 (short)0, acc3, false, false);
  }
  v8f acc = (acc0 + acc1) + (acc2 + acc3);

  // max over the 8 accumulator rows held by this lane, then across the wave
  float mx = acc[0];
#pragma unroll
  for (int i = 1; i < 8; ++i) mx = fmaxf(mx, acc[i]);
  for (int off = 16; off; off >>= 1) mx = fmaxf(mx, __shfl_xor(mx, off, 32));
  if (lane == 0) mc[(term * 16 + b) * 4 + t] = mx;

  if (t == 0) {
    for (int off = 16; off; off >>= 1) {
      sum_p2 += __shfl_xor(sum_p2, off, 32);
      sum_q2 += __shfl_xor(sum_q2, off, 32);
    }
    if (lane == 0) {
      ssq[(term * 16 + b) * 2 + 0] = sum_p2;
      ssq[(term * 16 + b) * 2 + 1] = sum_q2;
    }
  }
}

// ---------------------------------------------------------------------------
// Stage 3: combine everything into the scalar loss.
// ---------------------------------------------------------------------------
__global__ __launch_bounds__(256)
void mc_finalize(const float* __restrict__ partial,
                 const float* __restrict__ mc,
                 const float* __restrict__ ssq,
                 float* __restrict__ out) {
  __shared__ float red[5][256];
  int tid = threadIdx.x;
  for (int s = 0; s < 5; ++s)
    red[s][tid] = partial[s * NB_DICE + tid] + partial[s * NB_DICE + tid + 256];
  __syncthreads();
  for (int off = 128; off; off >>= 1) {
    if (tid < off) {
      for (int s = 0; s < 5; ++s) red[s][tid] += red[s][tid + off];
    }
    __syncthreads();
  }
  if (tid == 0) {
    float S_cg = red[0][0], S_c = red[1][0], S_g = red[2][0];
    float S_cs = red[3][0], S_s = red[4][0];
    const float SMOOTH = 1.0f, GAMMA = 0.5f;
    float dice1 = 1.f - (2.f * S_cg + SMOOTH) / (S_c + S_g + SMOOTH);
    float dice2 = 1.f - (2.f * S_cs + SMOOTH) / (S_c + S_s + SMOOTH);
    float tot[2];
    for (int term = 0; term < 2; ++term) {
      float acc = 0.f;
      for (int bb = 0; bb < 16; ++bb) {
        const float* m = &mc[(term * 16 + bb) * 4];
        float mx = fmaxf(fmaxf(m[0], m[1]), fmaxf(m[2], m[3]));
        float sp = ssq[(term * 16 + bb) * 2 + 0];
        float sq = ssq[(term * 16 + bb) * 2 + 1];
        // min_r MSE = (sum p^2 + sum q^2 - 2*max_r corr) / (N*2)
        acc += (sp + sq - 2.f * mx) * (1.f / 2048.f);
      }
      tot[term] = acc;
    }
    out[0] = dice1 + tot[0] + GAMMA * (dice2 + tot[1]);
  }
}

extern "C" void kernel_launch(void* const* d_in, const int* in_sizes, int n_in,
                              void* d_out, int out_size, void* d_ws, size_t ws_size,
                              hipStream_t stream) {
  const float* gt_mask         = (const float*)d_in[0];
  const float* gt_contour      = (const float*)d_in[1];
  const float* snake_gt_size   = (const float*)d_in[2];
  const float* snake_cl_size   = (const float*)d_in[3];
  const float* snake_mask      = (const float*)d_in[4];
  const float* classic_contour = (const float*)d_in[5];
  const float* classic_mask    = (const float*)d_in[6];

  float* ws      = (float*)d_ws;
  float* partial = ws;                    // 5 * NB_DICE floats
  float* mc      = ws + 5 * NB_DICE;      // 2*16*4 = 128 floats (max corr per tile)
  float* ssq     = mc + 128;              // 2*16*2 = 64 floats (sum p^2 / q^2)

  int n4 = in_sizes[0] / 4;               // masks as float4

  mc_dice_partials<<<NB_DICE, 256, 0, stream>>>(
      (const float4*)gt_mask, (const float4*)snake_mask,
      (const float4*)classic_mask, partial, n4);

  // 2 loss terms * 16 batches * 4 v-tiles = 128 one-wave workgroups
  mc_rot_corr<<<128, 32, 0, stream>>>(snake_gt_size, gt_contour,
                                      snake_cl_size, classic_contour,
                                      mc, ssq);

  mc_finalize<<<1, 256, 0, stream>>>(partial, mc, ssq, (float*)d_out);
}